// CifarResNet_86002425135605
// MI455X (gfx1250) — compile-verified
//
#include <hip/hip_runtime.h>
#include <hip/hip_bf16.h>
#include <cstddef>

typedef __attribute__((ext_vector_type(2))) float v2f;
typedef __attribute__((ext_vector_type(4))) float v4f;
typedef __attribute__((ext_vector_type(8))) float v8f;

// ============================================================================
// conv_dt: fused unfold + cdist + per-row standardize + exp scaling.
// One wave per (image b, 16-row L-tile); wave owns ALL M centers (MT tiles).
// GEMM via V_WMMA_F32_16X16X4_F32 (fp32-exact, matches fp32 reference).
// All shape params are template constants; OOB handling is branchless
// (address clamp + value select); center fragments load as b64; outputs
// store as b128 through an LDS transpose.
// ============================================================================
template<int MT, int K, int C, int H, int S>
__global__ __launch_bounds__(32)
void conv_dt_kernel(const float* __restrict__ x,        // (B,C,H,W)
                    const float* __restrict__ centers,  // (M,D)
                    const float* __restrict__ scaler,   // (M)
                    const float* __restrict__ biasp,    // (M)
                    float* __restrict__ out)            // (B,M,L)
{
    constexpr int M  = MT * 16;
    constexpr int W  = H;
    constexpr int P  = (K == 3) ? 1 : 0;
    constexpr int kk = K * K;
    constexpr int D  = C * kk;
    constexpr int D4 = (D + 3) & ~3;
    constexpr int oH = (H + 2 * P - K) / S + 1;
    constexpr int oW = oH;                 // power of two for all layers
    constexpr int L  = oH * oW;
    constexpr int tilesL = L / 16;

    const int lane = threadIdx.x;          // 0..31 (wave32)
    const int half = lane >> 4;            // 0 | 1
    const int lp   = lane & 15;
    const int b  = blockIdx.x / tilesL;    // tilesL compile-time -> cheap
    const int lt = blockIdx.x - b * tilesL;
    const int l0 = lt << 4;

    // A-matrix row handled by this lane (both halves cover same 16 rows,
    // different K: vgpr0/vgpr1 hold K = 2*half, 2*half+1).
    const int lA  = l0 + lp;
    const int oy  = lA / oW;               // shift (oW power of two)
    const int ox  = lA - oy * oW;
    const int iy0 = oy * S - P;
    const int ix0 = ox * S - P;
    const float* xb = x + (size_t)b * C * H * W;

    // Branchless unfold gather: clamp address, select value.
    auto fetchA = [&](int d) -> float {
        int c   = d / kk;                  // const divisor (9 or 1): magic mul
        int rem = d - c * kk;
        int i   = rem / K;                 // const divisor (3 or 1)
        int j   = rem - i * K;
        int yy = iy0 + i, xx = ix0 + j;
        bool ok = (d < D) && ((unsigned)yy < (unsigned)H)
                          && ((unsigned)xx < (unsigned)W);
        int off = (c * H + yy) * W + xx;
        off = ok ? off : 0;                // xb[0] always valid
        float v = xb[off];
        return ok ? v : 0.0f;
    };
    // Center pair: contiguous; one b64 load when D even (all layers but stem).
    auto fetchB2 = [&](int m, int d) -> v2f {
        if constexpr ((D % 2) == 0) {
            return *reinterpret_cast<const v2f*>(centers + m * D + d);
        } else {
            v2f r;
            bool ok0 = (d < D), ok1 = (d + 1 < D);
            float v0 = centers[m * D + (ok0 ? d : 0)];
            float v1 = centers[m * D + (ok1 ? d + 1 : 0)];
            r.x = ok0 ? v0 : 0.0f;
            r.y = ok1 ? v1 : 0.0f;
            return r;
        }
    };

    v8f acc[MT];
    float cn2[MT];
    v8f vzero = {};
    #pragma unroll
    for (int t = 0; t < MT; ++t) { acc[t] = vzero; cn2[t] = 0.0f; }
    float pn2 = 0.0f;

    constexpr int ITER = D4 / 4;
    constexpr int UMAX = (MT >= 4) ? 4 : 8;   // cap unroll: bound VGPR pressure
    constexpr int U = (UMAX >= 8 && ITER % 8 == 0) ? 8
                    : (ITER % 4 == 0) ? 4
                    : (ITER % 2 == 0) ? 2 : 1;

    for (int kb = 0; kb < D4; kb += 4 * U) {
        #pragma unroll
        for (int u = 0; u < U; ++u) {
            const int d0 = kb + 4 * u + 2 * half;
            v2f a;
            a.x = fetchA(d0);
            a.y = fetchA(d0 + 1);
            pn2 += a.x * a.x + a.y * a.y;   // per-lane partial of ||patch||^2

            #pragma unroll
            for (int t = 0; t < MT; ++t) {
                const int m = (t << 4) + lp;          // B-frag: N = lane%16
                v2f bf = fetchB2(m, d0);
                cn2[t] += bf.x * bf.x + bf.y * bf.y;  // partial ||center||^2
                acc[t] = __builtin_amdgcn_wmma_f32_16x16x4_f32(
                             false, a, false, bf, (short)0, acc[t], false, false);
            }
        }
    }

    // Finish ||patch||^2 (halves cover disjoint K) -> per-row value in LDS.
    pn2 += __shfl_xor(pn2, 16);
    __shared__ float pnS[16];
    __shared__ float tileS[16 * M];
    if (lane < 16) pnS[lane] = pn2;
    __syncthreads();

    float scl[MT], bia[MT];
    #pragma unroll
    for (int t = 0; t < MT; ++t) {
        cn2[t] += __shfl_xor(cn2[t], 16);   // ||center_m||^2, m = t*16+lp
        scl[t] = scaler[(t << 4) + lp];
        bia[t] = biasp[(t << 4) + lp];
    }

    // d = sqrt(max(||p||^2 + ||c||^2 - 2 p.c, 0)); C-layout row = i + 8*half.
    float dv[MT][8];
    #pragma unroll
    for (int t = 0; t < MT; ++t)
        #pragma unroll
        for (int i = 0; i < 8; ++i) {
            int r = i + 8 * half;
            float d2 = pnS[r] + cn2[t] - 2.0f * acc[t][i];
            dv[t][i] = sqrtf(fmaxf(d2, 0.0f));
        }

    // Standardize over M centers (row lives across 16 lanes of a half-wave).
    #pragma unroll
    for (int i = 0; i < 8; ++i) {
        float s = 0.0f, ss = 0.0f;
        #pragma unroll
        for (int t = 0; t < MT; ++t) { s += dv[t][i]; ss += dv[t][i] * dv[t][i]; }
        #pragma unroll
        for (int mk = 1; mk < 16; mk <<= 1) {
            s  += __shfl_xor(s,  mk);
            ss += __shfl_xor(ss, mk);
        }
        const float mean = s * (1.0f / M);
        const float var  = (ss - s * s * (1.0f / M)) * (1.0f / (M - 1)); // ddof=1
        const float rstd = rsqrtf(var);
        #pragma unroll
        for (int t = 0; t < MT; ++t) {
            float dn = (dv[t][i] - mean) * rstd;
            dv[t][i] = expf((-dn - 3.0f) * scl[t]) + bia[t];
        }
    }

    // LDS transpose -> vectorized (b128) coalesced stores along L per channel.
    #pragma unroll
    for (int t = 0; t < MT; ++t)
        #pragma unroll
        for (int i = 0; i < 8; ++i) {
            int r = i + 8 * half;
            tileS[((t << 4) + lp) * 16 + r] = dv[t][i];
        }
    __syncthreads();
    float* ob = out + (size_t)b * M * L + l0;
    constexpr int NJ = (16 * M) / (4 * 32);     // float4 chunks per lane
    #pragma unroll
    for (int j = 0; j < NJ; ++j) {
        int ch = lane + 32 * j;
        int m  = ch >> 2;                        // 4 chunks per channel row
        int r0 = (ch & 3) << 2;
        v4f val = *reinterpret_cast<const v4f*>(&tileS[m * 16 + r0]);
        *reinterpret_cast<v4f*>(&ob[(size_t)m * L + r0]) = val;
    }
}

// ============================================================================
// BatchNorm (training mode, batch statistics, biased var) — stats pass.
// L and M are powers of two: shift/mask addressing (lshift = log2 L).
// ============================================================================
__global__ __launch_bounds__(256)
void bn_stats_kernel(const float* __restrict__ x, float* __restrict__ stats,
                     int B, int M, int lshift)
{
    const int m = blockIdx.x;
    const int tid = threadIdx.x;
    const int L = 1 << lshift;
    const long long N = (long long)B << lshift;
    float s = 0.0f, ss = 0.0f;
    for (long long i = tid; i < N; i += 256) {
        int bb = (int)(i >> lshift);
        int l  = (int)i & (L - 1);
        float v = x[(((size_t)bb * M + m) << lshift) + l];
        s += v; ss += v * v;
    }
    __shared__ float rs[256], rss[256];
    rs[tid] = s; rss[tid] = ss;
    __syncthreads();
    for (int off = 128; off > 0; off >>= 1) {
        if (tid < off) { rs[tid] += rs[tid + off]; rss[tid] += rss[tid + off]; }
        __syncthreads();
    }
    if (tid == 0) {
        float mean = rs[0] / (float)N;
        float var  = rss[0] / (float)N - mean * mean;
        stats[2 * m]     = mean;
        stats[2 * m + 1] = rsqrtf(fmaxf(var, 0.0f) + 1e-5f);
    }
}

// BN apply (+ optional residual, + optional ReLU), in place.
__global__ __launch_bounds__(256)
void bn_apply_kernel(float* __restrict__ x, const float* __restrict__ stats,
                     const float* __restrict__ g, const float* __restrict__ bb,
                     const float* __restrict__ residual,
                     int M, int lshift, long long total, int relu)
{
    long long i = (long long)blockIdx.x * blockDim.x + threadIdx.x;
    if (i >= total) return;
    int m = (int)(i >> lshift) & (M - 1);     // M power of two
    float v = (x[i] - stats[2 * m]) * stats[2 * m + 1] * g[m] + bb[m];
    if (residual) v += residual[i];
    if (relu) v = fmaxf(v, 0.0f);
    x[i] = v;
}

// Global average pool over L per (b, m).
__global__ void pool_kernel(const float* __restrict__ x, float* __restrict__ p,
                            int M, int L)
{
    int b = blockIdx.x, m = threadIdx.x;
    const float* xp = x + ((size_t)b * M + m) * L;
    float s = 0.0f;
    for (int l = 0; l < L; ++l) s += xp[l];
    p[b * M + m] = s / (float)L;
}

// Final FC: (B,64) x (10,64)^T + bias.
__global__ void fc_kernel(const float* __restrict__ p, const float* __restrict__ w,
                          const float* __restrict__ bb, float* __restrict__ out,
                          int B, int K, int N)
{
    int i = blockIdx.x * blockDim.x + threadIdx.x;
    if (i >= B * N) return;
    int b = i / N, n = i - b * N;
    float s = bb[n];
    for (int kq = 0; kq < K; ++kq) s += p[b * K + kq] * w[n * K + kq];
    out[i] = s;
}

// ============================================================================
// Host orchestration
// ============================================================================
namespace {
struct DTp { const float *c, *s, *b; };
struct BNp { const float *g, *b; };
struct Blk { DTp c1; BNp b1; DTp c2; BNp b2; bool ds; DTp dc; BNp db; };
constexpr int BATCH = 256;

constexpr int ctz_c(int v) { int s = 0; while (!(v & 1)) { v >>= 1; ++s; } return s; }

template<int MT, int K, int C, int H, int S>
void launch_conv(const float* in, float* outb, const DTp& dt, hipStream_t stream)
{
    constexpr int P  = (K == 3) ? 1 : 0;
    constexpr int OH = (H + 2 * P - K) / S + 1;
    constexpr int L  = OH * OH;
    dim3 grid((unsigned)(BATCH * (L / 16))), blk(32);
    conv_dt_kernel<MT, K, C, H, S><<<grid, blk, 0, stream>>>(
        in, dt.c, dt.s, dt.b, outb);
}

void launch_bn(float* buf, const BNp& bn, float* stats, int M, int lshift,
               const float* res, int relu, hipStream_t stream)
{
    bn_stats_kernel<<<M, 256, 0, stream>>>(buf, stats, BATCH, M, lshift);
    long long tot = ((long long)BATCH * M) << lshift;
    bn_apply_kernel<<<(unsigned)((tot + 255) / 256), 256, 0, stream>>>(
        buf, stats, bn.g, bn.b, res, M, lshift, tot, relu);
}

template<int C, int H, int S, int M, bool DS>
void run_block(Blk& bp, float*& cur, float* const* P, float* stats,
               hipStream_t stream)
{
    constexpr int OH  = (H + 2 - 3) / S + 1;
    constexpr int Lo  = OH * OH;
    constexpr int lsh = ctz_c(Lo);
    float *t1, *t2;
    if (cur == P[0])      { t1 = P[1]; t2 = P[2]; }
    else if (cur == P[1]) { t1 = P[0]; t2 = P[2]; }
    else                  { t1 = P[0]; t2 = P[1]; }

    launch_conv<M / 16, 3, C, H, S>(cur, t1, bp.c1, stream);
    launch_bn(t1, bp.b1, stats, M, lsh, nullptr, 1, stream);
    launch_conv<M / 16, 3, M, OH, 1>(t1, t2, bp.c2, stream);
    const float* idn = cur;
    if (DS) {
        launch_conv<M / 16, 1, C, H, S>(cur, P[3], bp.dc, stream);
        launch_bn(P[3], bp.db, stats, M, lsh, nullptr, 0, stream);
        idn = P[3];
    }
    launch_bn(t2, bp.b2, stats, M, lsh, idn, 1, stream);
    cur = t2;
}
} // namespace

extern "C" void kernel_launch(void* const* d_in, const int* in_sizes, int n_in,
                              void* d_out, int out_size, void* d_ws, size_t ws_size,
                              hipStream_t stream)
{
    (void)in_sizes; (void)n_in; (void)out_size; (void)ws_size;
    const float* x = (const float*)d_in[0];
    int idx = 1;
    auto F     = [&]() { return (const float*)d_in[idx++]; };
    auto getDT = [&]() { DTp p; p.c = F(); p.s = F(); p.b = F(); return p; };
    auto getBN = [&]() { BNp p; p.g = F(); p.b = F(); return p; };
    auto getBlk = [&](bool ds) {
        Blk b;
        b.c1 = getDT(); b.b1 = getBN();
        b.c2 = getDT(); b.b2 = getBN();
        b.ds = ds;
        if (ds) { b.dc = getDT(); b.db = getBN(); }
        return b;
    };

    DTp conv1 = getDT(); BNp bn1 = getBN();
    Blk l1[3], l2[3], l3[3];
    for (int i = 0; i < 3; ++i) l1[i] = getBlk(false);
    l2[0] = getBlk(true);  l2[1] = getBlk(false); l2[2] = getBlk(false);
    l3[0] = getBlk(true);  l3[1] = getBlk(false); l3[2] = getBlk(false);
    const float* fcw = F();
    const float* fcb = F();

    float* ws = (float*)d_ws;
    const size_t BUF = (size_t)BATCH * 16 * 1024;   // 4M floats = 16 MB (max act)
    float* P[4] = { ws, ws + BUF, ws + 2 * BUF, ws + 3 * BUF };
    float* stats  = ws + 4 * BUF;                   // 128 floats
    float* pooled = stats + 256;                    // 256*64 floats

    // Stem: C=3, H=32, k=3, s=1, M=16
    launch_conv<1, 3, 3, 32, 1>(x, P[0], conv1, stream);
    launch_bn(P[0], bn1, stats, 16, 10, nullptr, 1, stream);
    float* cur = P[0];

    // layer1: 3 x block(16 -> 16, H=32, s=1)
    run_block<16, 32, 1, 16, false>(l1[0], cur, P, stats, stream);
    run_block<16, 32, 1, 16, false>(l1[1], cur, P, stats, stream);
    run_block<16, 32, 1, 16, false>(l1[2], cur, P, stats, stream);
    // layer2: block(16 -> 32, s=2, ds) + 2 x block(32 -> 32, H=16)
    run_block<16, 32, 2, 32, true >(l2[0], cur, P, stats, stream);
    run_block<32, 16, 1, 32, false>(l2[1], cur, P, stats, stream);
    run_block<32, 16, 1, 32, false>(l2[2], cur, P, stats, stream);
    // layer3: block(32 -> 64, s=2, ds) + 2 x block(64 -> 64, H=8)
    run_block<32, 16, 2, 64, true >(l3[0], cur, P, stats, stream);
    run_block<64,  8, 1, 64, false>(l3[1], cur, P, stats, stream);
    run_block<64,  8, 1, 64, false>(l3[2], cur, P, stats, stream);

    pool_kernel<<<BATCH, 64, 0, stream>>>(cur, pooled, 64, 64);
    fc_kernel<<<(BATCH * 10 + 255) / 256, 256, 0, stream>>>(
        pooled, fcw, fcb, (float*)d_out, BATCH, 64, 10);
}